// SheafHyperGNN_8160437862724
// MI455X (gfx1250) — compile-verified
//
#include <hip/hip_runtime.h>
#include <math.h>

// ---------------- problem constants (match reference) ----------------
constexpr int N_NODES = 50000;
constexpr int N_EDGES = 10000;
constexpr int NNZ     = 150000;
constexpr int IN_CH   = 128;
constexpr int HIDDEN  = 64;
constexpr int D       = 6;
constexpr int OUT_CH  = HIDDEN * D;          // 384
constexpr int ND      = N_NODES * D;         // 300000
constexpr int ED      = N_EDGES * D;         // 60000

typedef __attribute__((ext_vector_type(2))) float v2f;
typedef __attribute__((ext_vector_type(8))) float v8f;

// ---------------------------------------------------------------------
// Full-precision f32 WMMA GEMM: C[M,N] = A[M,K] @ B[K,N] (+ bias[N]).
// One 16x16 output tile per wave; A/B streamed from L2 (x, W_lin, W_conv
// all fit in the 192MB L2, so no LDS staging is needed).
// A-frag (16x4 f32, 2 VGPR): lanes 0-15 -> K={k0,k0+1}, lanes 16-31 -> K={k0+2,k0+3}
// B-frag (4x16 f32, 2 VGPR): mirrored; C/D: VGPR j -> row j + 8*(lane>=16).
// ---------------------------------------------------------------------
__global__ void wmma_gemm_f32(const float* __restrict__ A, const float* __restrict__ B,
                              const float* __restrict__ bias, float* __restrict__ C,
                              int K, int N, int nTiles, int totalTiles) {
  int tile = blockIdx.x * (blockDim.x >> 5) + (threadIdx.x >> 5);
  if (tile >= totalTiles) return;            // uniform per wave: EXEC stays all-ones
  int lane = threadIdx.x & 31;
  int half = lane >> 4;                      // 0: lanes 0-15, 1: lanes 16-31
  int ml   = lane & 15;                      // A row within tile / B-C column within tile
  int m0 = (tile / nTiles) * 16;
  int n0 = (tile % nTiles) * 16;

  v8f acc = {};
  const float* Arow = A + (size_t)(m0 + ml) * K + 2 * half;
  const float* Bcol = B + (size_t)(2 * half) * N + n0 + ml;
  for (int k0 = 0; k0 < K; k0 += 4) {
    v2f a, b;
    a.x = Arow[k0];
    a.y = Arow[k0 + 1];
    b.x = Bcol[(size_t)k0 * N];
    b.y = Bcol[(size_t)(k0 + 1) * N];
    acc = __builtin_amdgcn_wmma_f32_16x16x4_f32(false, a, false, b,
                                                (short)0, acc, false, false);
  }
  float bv = bias ? bias[n0 + ml] : 0.0f;
#pragma unroll
  for (int j = 0; j < 8; ++j) {
    C[(size_t)(m0 + j + 8 * half) * N + n0 + ml] = acc[j] + bv;
  }
}

// ---------------------------------------------------------------------
// mean over D stalks: m[n,h] = (1/6) * sum_t proj[n, t*64 + h]
// ---------------------------------------------------------------------
__global__ void stalk_mean_kernel(const float* __restrict__ proj, float* __restrict__ m,
                                  int total) {
  int idx = blockIdx.x * blockDim.x + threadIdx.x;
  if (idx >= total) return;
  int n = idx >> 6;
  int hh = idx & 63;
  const float* p = proj + (size_t)n * OUT_CH + hh;
  float s = 0.0f;
#pragma unroll
  for (int t = 0; t < D; ++t) s += p[t * HIDDEN];
  m[idx] = s * (1.0f / 6.0f);
}

// butterfly all-reduce across wave32
__device__ inline float wave_allsum(float v) {
#pragma unroll
  for (int off = 16; off > 0; off >>= 1) v += __shfl_xor(v, off, 32);
  return v;
}

// ---------------------------------------------------------------------
// Sheaf map: one wave per incidence. Gather concat(xm[row], em[col]) (128),
// LayerNorm, @ W_sheaf (128x6) + b, sigmoid -> alpha[i,0..5].
// Also accumulates D_v = sum alpha^2 (per node-stalk) and hyperedge degree.
// ---------------------------------------------------------------------
__global__ void alpha_kernel(const float* __restrict__ xm, const float* __restrict__ em,
                             const int* __restrict__ row, const int* __restrict__ col,
                             const float* __restrict__ ln_g, const float* __restrict__ ln_b,
                             const float* __restrict__ Wsh, const float* __restrict__ bsh,
                             float* __restrict__ alpha, float* __restrict__ Dsum,
                             float* __restrict__ Bcnt) {
  int i = blockIdx.x * (blockDim.x >> 5) + (threadIdx.x >> 5);
  if (i >= NNZ) return;
  int lane = threadIdx.x & 31;
  int rn = row[i], cn = col[i];
  int j0 = lane * 4;                        // 4 of 128 feature elements per lane
  float z[4];
#pragma unroll
  for (int q = 0; q < 4; ++q) {
    int j = j0 + q;
    z[q] = (j < HIDDEN) ? xm[(size_t)rn * HIDDEN + j]
                        : em[(size_t)cn * HIDDEN + (j - HIDDEN)];
  }
  float s  = z[0] + z[1] + z[2] + z[3];
  float ss = z[0]*z[0] + z[1]*z[1] + z[2]*z[2] + z[3]*z[3];
  s  = wave_allsum(s);
  ss = wave_allsum(ss);
  float mu   = s * (1.0f / 128.0f);
  float var  = ss * (1.0f / 128.0f) - mu * mu;
  float rstd = rsqrtf(var + 1e-5f);

  float p[6] = {0, 0, 0, 0, 0, 0};
#pragma unroll
  for (int q = 0; q < 4; ++q) {
    int j = j0 + q;
    float fn = (z[q] - mu) * rstd * ln_g[j] + ln_b[j];
#pragma unroll
    for (int t = 0; t < D; ++t) p[t] += fn * Wsh[j * D + t];
  }
#pragma unroll
  for (int t = 0; t < D; ++t) p[t] = wave_allsum(p[t]);

  if (lane < D) {
    float a = 1.0f / (1.0f + expf(-(p[lane] + bsh[lane])));
    alpha[(size_t)i * D + lane] = a;
    atomicAdd(&Dsum[rn * D + lane], a * a);
  } else if (lane == D) {
    atomicAdd(&Bcnt[cn], 1.0f);
  }
}

__global__ void safe_inv_kernel(float* __restrict__ buf, int n) {
  int idx = blockIdx.x * blockDim.x + threadIdx.x;
  if (idx >= n) return;
  float v = buf[idx];
  buf[idx] = (v > 0.0f) ? (1.0f / v) : 0.0f;
}

// node-stalk -> hyperedge-stalk: me[col*6+t,:] += alpha[i,t] * h[row*6+t,:]
__global__ void scatter_me_kernel(const float* __restrict__ h, const float* __restrict__ alpha,
                                  const int* __restrict__ row, const int* __restrict__ col,
                                  float* __restrict__ me) {
  int idx = blockIdx.x * blockDim.x + threadIdx.x;
  if (idx >= NNZ * OUT_CH) return;
  int i  = idx / OUT_CH;
  int r  = idx - i * OUT_CH;
  int t  = r >> 6;
  int hh = r & 63;
  float a = alpha[(size_t)i * D + t];
  float v = a * h[(size_t)(row[i] * D + t) * HIDDEN + hh];
  atomicAdd(&me[(size_t)(col[i] * D + t) * HIDDEN + hh], v);
}

// hyperedge-stalk -> node-stalk: agg[row*6+t,:] += alpha[i,t]*Binv[col]*me[col*6+t,:]
__global__ void scatter_agg_kernel(const float* __restrict__ me, const float* __restrict__ alpha,
                                   const int* __restrict__ row, const int* __restrict__ col,
                                   const float* __restrict__ Binv, float* __restrict__ agg) {
  int idx = blockIdx.x * blockDim.x + threadIdx.x;
  if (idx >= NNZ * OUT_CH) return;
  int i  = idx / OUT_CH;
  int r  = idx - i * OUT_CH;
  int t  = r >> 6;
  int hh = r & 63;
  int cn = col[i];
  float a = alpha[(size_t)i * D + t] * Binv[cn];
  float v = a * me[(size_t)(cn * D + t) * HIDDEN + hh];
  atomicAdd(&agg[(size_t)(row[i] * D + t) * HIDDEN + hh], v);
}

// out = elu( h - Dinv[r]*agg + bias[h] )   (elu after both convs per reference)
__global__ void combine_kernel(const float* __restrict__ h, const float* __restrict__ agg,
                               const float* __restrict__ Dinv, const float* __restrict__ bias,
                               float* __restrict__ out, int total) {
  int idx = blockIdx.x * blockDim.x + threadIdx.x;
  if (idx >= total) return;
  int r  = idx >> 6;
  int hh = idx & 63;
  float o = h[idx] - Dinv[r] * agg[idx] + bias[hh];
  out[idx] = (o > 0.0f) ? o : expm1f(o);
}

// ---------------------------------------------------------------------
extern "C" void kernel_launch(void* const* d_in, const int* in_sizes, int n_in,
                              void* d_out, int out_size, void* d_ws, size_t ws_size,
                              hipStream_t stream) {
  (void)in_sizes; (void)n_in; (void)out_size; (void)ws_size;
  const float* x       = (const float*)d_in[0];
  const float* hattr   = (const float*)d_in[1];
  const float* W_lin   = (const float*)d_in[2];
  const float* b_lin   = (const float*)d_in[3];
  const float* ln_g    = (const float*)d_in[4];
  const float* ln_b    = (const float*)d_in[5];
  const float* W_sheaf = (const float*)d_in[6];
  const float* b_sheaf = (const float*)d_in[7];
  const float* W_conv  = (const float*)d_in[8];   // (2,64,64)
  const float* b_conv  = (const float*)d_in[9];   // (2,64)
  const int*   eidx    = (const int*)d_in[10];
  const int* row = eidx;
  const int* col = eidx + NNZ;
  float* out = (float*)d_out;                     // (50000, 384) == (300000, 64)

  // ---- workspace layout (floats) ----
  float* ws    = (float*)d_ws;
  float* xs    = ws;                      // 300000*64 = 19,200,000
  float* es    = xs   + (size_t)ND * HIDDEN;        // 3,840,000
  float* xm    = es   + (size_t)ED * HIDDEN;        // 3,200,000
  float* em    = xm   + (size_t)N_NODES * HIDDEN;   //   640,000
  float* alpha = em   + (size_t)N_EDGES * HIDDEN;   //   900,000
  float* Dbuf  = alpha + (size_t)NNZ * D;           //   300,000 (Dsum -> Dinv in place)
  float* Bbuf  = Dbuf + ND;                         //    10,000 (deg -> 1/deg in place)
  float* hbuf  = Bbuf + N_EDGES;                    // 19,200,000
  float* me    = hbuf + (size_t)ND * HIDDEN;        // 3,840,000
  float* agg   = me   + (size_t)ED * HIDDEN;        // 19,200,000

  const int TPB = 256;

  // 1) input projections (fp32 WMMA GEMM, bias fused): xs = x@W_lin+b, es = he@W_lin+b
  {
    int tiles = (N_NODES / 16) * (OUT_CH / 16);     // 3125*24 = 75000
    wmma_gemm_f32<<<tiles / 8, TPB, 0, stream>>>(x, W_lin, b_lin, xs,
                                                 IN_CH, OUT_CH, OUT_CH / 16, tiles);
    int etiles = (N_EDGES / 16) * (OUT_CH / 16);    // 625*24 = 15000
    wmma_gemm_f32<<<etiles / 8, TPB, 0, stream>>>(hattr, W_lin, b_lin, es,
                                                  IN_CH, OUT_CH, OUT_CH / 16, etiles);
  }

  // 2) stalk means
  stalk_mean_kernel<<<(N_NODES * HIDDEN + TPB - 1) / TPB, TPB, 0, stream>>>(xs, xm, N_NODES * HIDDEN);
  stalk_mean_kernel<<<(N_EDGES * HIDDEN + TPB - 1) / TPB, TPB, 0, stream>>>(es, em, N_EDGES * HIDDEN);

  // 3) sheaf coefficients + degrees
  hipMemsetAsync(Dbuf, 0, (size_t)ND * sizeof(float), stream);
  hipMemsetAsync(Bbuf, 0, (size_t)N_EDGES * sizeof(float), stream);
  alpha_kernel<<<NNZ / 8, TPB, 0, stream>>>(xm, em, row, col, ln_g, ln_b,
                                            W_sheaf, b_sheaf, alpha, Dbuf, Bbuf);
  safe_inv_kernel<<<(ND + TPB - 1) / TPB, TPB, 0, stream>>>(Dbuf, ND);
  safe_inv_kernel<<<(N_EDGES + TPB - 1) / TPB, TPB, 0, stream>>>(Bbuf, N_EDGES);

  const int convTiles  = (ND / 16) * (HIDDEN / 16);  // 18750*4 = 75000
  const int scatterBlk = (NNZ * OUT_CH + TPB - 1) / TPB;
  const int elemBlk    = (ND * HIDDEN + TPB - 1) / TPB;

  // 4) conv layer 1: x1 = elu( h1 - Dinv * H^T Binv H h1 + b0 ),  h1 = xs @ W0
  wmma_gemm_f32<<<convTiles / 8, TPB, 0, stream>>>(xs, W_conv, nullptr, hbuf,
                                                   HIDDEN, HIDDEN, HIDDEN / 16, convTiles);
  hipMemsetAsync(me, 0, (size_t)ED * HIDDEN * sizeof(float), stream);
  scatter_me_kernel<<<scatterBlk, TPB, 0, stream>>>(hbuf, alpha, row, col, me);
  hipMemsetAsync(agg, 0, (size_t)ND * HIDDEN * sizeof(float), stream);
  scatter_agg_kernel<<<scatterBlk, TPB, 0, stream>>>(me, alpha, row, col, Bbuf, agg);
  combine_kernel<<<elemBlk, TPB, 0, stream>>>(hbuf, agg, Dbuf, b_conv, out, ND * HIDDEN); // x1 -> d_out

  // 5) conv layer 2: out = elu( h2 - Dinv * H^T Binv H h2 + b1 ),  h2 = x1 @ W1
  wmma_gemm_f32<<<convTiles / 8, TPB, 0, stream>>>(out, W_conv + HIDDEN * HIDDEN, nullptr, hbuf,
                                                   HIDDEN, HIDDEN, HIDDEN / 16, convTiles);
  hipMemsetAsync(me, 0, (size_t)ED * HIDDEN * sizeof(float), stream);
  scatter_me_kernel<<<scatterBlk, TPB, 0, stream>>>(hbuf, alpha, row, col, me);
  hipMemsetAsync(agg, 0, (size_t)ND * HIDDEN * sizeof(float), stream);
  scatter_agg_kernel<<<scatterBlk, TPB, 0, stream>>>(me, alpha, row, col, Bbuf, agg);
  combine_kernel<<<elemBlk, TPB, 0, stream>>>(hbuf, agg, Dbuf, b_conv + HIDDEN, out, ND * HIDDEN);
}